// PrimitiveCollisionCost_19533511262442
// MI455X (gfx1250) — compile-verified
//
#include <hip/hip_runtime.h>
#include <math.h>

// Problem dims (fixed at compile time, from the reference)
#define BQ 512
#define HQ 32
#define LQ 7
#define SQ 5
#define OQ 8
#define NQ (BQ * HQ)            // 16384 poses
#define NPB 16                  // poses per block
#define ROWS (NPB * LQ * SQ)    // 560 sphere rows per block = 35 exact 16-row tiles
#define THREADS 224             // 7 waves; 5 WMMA tiles per wave
#define TILES (ROWS / 16)       // 35
#define TPW (TILES / 7)         // 5 tiles per wave

typedef __attribute__((ext_vector_type(2))) float v2f;
typedef __attribute__((ext_vector_type(4))) float v4f;
typedef __attribute__((ext_vector_type(8))) float v8f;

__global__ __launch_bounds__(THREADS)
void collision_cost_kernel(const float* __restrict__ pos,    // [N,L,3]
                           const float* __restrict__ rot,    // [N,L,3,3]
                           const float* __restrict__ soff,   // [L,S,3]
                           const float* __restrict__ srad,   // [L,S]
                           const float* __restrict__ ocent,  // [O,3]
                           const float* __restrict__ orad,   // [O]
                           const float* __restrict__ weight, // [1]
                           float* __restrict__ out)          // [N]
{
    __shared__ __align__(16) float rotL[NPB * LQ * 9];   // 1008
    __shared__ __align__(16) float posL[NPB * LQ * 3];   // 336
    __shared__ float offL[LQ * SQ * 3];                  // 105
    __shared__ float sradL[LQ * SQ];                     // 35
    __shared__ float ocentL[OQ * 3];                     // 24
    __shared__ float oradL[OQ];                          // 8
    __shared__ __align__(16) float As[ROWS * 4];         // center rows: [cx,cy,cz,|c|^2]
    __shared__ float rowmax[ROWS];                       // max over objects per sphere-row

    const int tid = threadIdx.x;
    const int blk = blockIdx.x;

    // ---- Phase 0: stage the streaming inputs into LDS (coalesced b128) ----
    {
        const v4f* rg = (const v4f*)(rot + (size_t)blk * (NPB * LQ * 9));
        v4f* rl = (v4f*)rotL;
        for (int i = tid; i < (NPB * LQ * 9) / 4; i += THREADS) rl[i] = rg[i];
        const v4f* pg = (const v4f*)(pos + (size_t)blk * (NPB * LQ * 3));
        v4f* pl = (v4f*)posL;
        for (int i = tid; i < (NPB * LQ * 3) / 4; i += THREADS) pl[i] = pg[i];
        for (int i = tid; i < LQ * SQ * 3; i += THREADS) offL[i] = soff[i];
        for (int i = tid; i < LQ * SQ; i += THREADS) sradL[i] = srad[i];
        if (tid < OQ * 3) ocentL[tid] = ocent[tid];
        if (tid < OQ)     oradL[tid]  = orad[tid];
    }
    __syncthreads();

    // ---- Phase 1: rotate link-frame spheres into world frame; build B columns ----
    for (int r = tid; r < ROWS; r += THREADS) {
        const int nl  = r / (LQ * SQ);
        const int rem = r - nl * (LQ * SQ);
        const int l   = rem / SQ;
        const int s   = rem - l * SQ;
        const float* R  = &rotL[(nl * LQ + l) * 9];
        const float* P  = &posL[(nl * LQ + l) * 3];
        const float* Ov = &offL[(l * SQ + s) * 3];
        const float ox = Ov[0], oy = Ov[1], oz = Ov[2];
        const float cx = R[0]*ox + R[1]*oy + R[2]*oz + P[0];
        const float cy = R[3]*ox + R[4]*oy + R[5]*oz + P[1];
        const float cz = R[6]*ox + R[7]*oy + R[8]*oz + P[2];
        v4f row; row.x = cx; row.y = cy; row.z = cz; row.w = cx*cx + cy*cy + cz*cz;
        *(v4f*)&As[r * 4] = row;                         // ds_store_b128
    }
    __syncthreads();

    // ---- Phase 2: transposed WMMA tiles  D[o,p] = |w_o|^2 - 2 w_o.c_p + |c_p|^2 ----
    // M = objects (rows 0..7 valid, 8..15 zero-padded), N = sphere rows, K = 4.
    const int lane = tid & 31;
    const int wave = tid >> 5;
    const int half = lane >> 4;   // K-slice selector (A and B), M-half for D
    const int col  = lane & 15;   // sphere-row column within tile

    // Loop-invariant per-lane constants: object radii and |w|^2 (C rows, M = vgpr idx).
    float oradr[OQ], onorm[OQ];
#pragma unroll
    for (int o = 0; o < OQ; ++o) {
        const float wx = ocentL[o*3+0], wy = ocentL[o*3+1], wz = ocentL[o*3+2];
        onorm[o] = wx*wx + wy*wy + wz*wz;
        oradr[o] = oradL[o];
    }
    // A (16x4 object matrix): lane holds row M=col, K pair 2*half..2*half+1.
    v2f avec; avec.x = 0.0f; avec.y = 0.0f;
    if (col < OQ) {
        const float wx = ocentL[col*3+0], wy = ocentL[col*3+1], wz = ocentL[col*3+2];
        if (half == 0) { avec.x = -2.0f * wx; avec.y = -2.0f * wy; }
        else           { avec.x = -2.0f * wz; avec.y = 1.0f;       }
    }
    // C: vgpr r holds M = r + 8*half -> |w_r|^2 for low half, pad 0 for high half.
    v8f cvec;
#pragma unroll
    for (int r = 0; r < 8; ++r) cvec[r] = (half == 0) ? onorm[r] : 0.0f;

    // Software pipeline: issue all 5 B-slice loads + 5 link-sphere radii up front
    // so ds latency overlaps the sqrt/max chains (waits become dscnt<=k, not 0).
    v2f  bv[TPW];
    float srv[TPW];
#pragma unroll
    for (int u = 0; u < TPW; ++u) {
        const int p = (wave * TPW + u) * 16 + col;
        bv[u] = *(const v2f*)&As[p * 4 + 2 * half];       // ds_load_b64
        const int rem = p % (LQ * SQ);
        const int l   = rem / SQ;
        const int s   = rem - l * SQ;
        srv[u] = sradL[l * SQ + s];                       // ds_load_b32
    }

#pragma unroll
    for (int u = 0; u < TPW; ++u) {
        v8f d = __builtin_amdgcn_wmma_f32_16x16x4_f32(
                    false, avec, false, bv[u], (short)0, cvec, false, false);
        if (half == 0) {
            // Lane owns all 8 object distances of sphere-row p in its D registers.
            const int p = (wave * TPW + u) * 16 + col;
            float m = -1.0e30f;
#pragma unroll
            for (int r = 0; r < OQ; ++r)
                m = fmaxf(m, (srv[u] + oradr[r]) - __builtin_amdgcn_sqrtf(d[r]));
            rowmax[p] = m;
        }
    }
    __syncthreads();

    // ---- Phase 3: max over spheres, clamp-scale, sum over links, weight ----
    if (tid < NPB) {
        float acc = 0.0f;
#pragma unroll
        for (int l = 0; l < LQ; ++l) {
            float mx = -1.0e30f;
#pragma unroll
            for (int s = 0; s < SQ; ++s)
                mx = fmaxf(mx, rowmax[tid * (LQ * SQ) + l * SQ + s]);
            acc += fminf(fmaxf(mx + 0.1f, 0.0f), 0.2f) * 4.0f;  // clip(..,0,0.2)/0.25
        }
        out[(size_t)blk * NPB + tid] = acc * weight[0];
    }
}

extern "C" void kernel_launch(void* const* d_in, const int* in_sizes, int n_in,
                              void* d_out, int out_size, void* d_ws, size_t ws_size,
                              hipStream_t stream) {
    const float* pos    = (const float*)d_in[0];
    const float* rot    = (const float*)d_in[1];
    const float* soff   = (const float*)d_in[2];
    const float* srad   = (const float*)d_in[3];
    const float* ocent  = (const float*)d_in[4];
    const float* orad   = (const float*)d_in[5];
    const float* weight = (const float*)d_in[6];
    float* out = (float*)d_out;

    dim3 grid(NQ / NPB);   // 1024 blocks of 16 poses
    dim3 block(THREADS);   // 7 waves
    collision_cost_kernel<<<grid, block, 0, stream>>>(
        pos, rot, soff, srad, ocent, orad, weight, out);
}